// DeformConv3d_12764642803824
// MI455X (gfx1250) — compile-verified
//
#include <hip/hip_runtime.h>

#define Bn   8
#define Cc   64
#define Hh   128
#define Ww   128
#define HW   (Hh * Ww)      // 16384
#define KKk  9
#define KC   (KKk * Cc)     // 576
#define NOFF 18
#define TP   16             // output pixels per workgroup

typedef __attribute__((ext_vector_type(2))) float v2f;
typedef __attribute__((ext_vector_type(8))) float v8f;

// ---------------------------------------------------------------------------
// Kernel 1: offsets = conv3x3(x, w_off) + b_off   (64 -> 18 channels, pad 1)
// thread = (b, ch, hw) flat; consecutive threads share (b,ch) -> coalesced x,
// scalar-broadcast weights.
// ---------------------------------------------------------------------------
__global__ __launch_bounds__(256) void offset_conv_kernel(
    const float* __restrict__ x, const float* __restrict__ w_off,
    const float* __restrict__ b_off, float* __restrict__ offs)
{
    int idx = blockIdx.x * blockDim.x + threadIdx.x;   // b*NOFF*HW + ch*HW + hw
    int hw  = idx & (HW - 1);
    int t   = idx >> 14;                               // / HW
    int ch  = t % NOFF;
    int b   = t / NOFF;
    int h = hw >> 7, w = hw & 127;

    float acc = b_off[ch];
    const float* xb = x + (size_t)b * Cc * HW;
    const float* wc = w_off + ch * Cc * 9;

    for (int c = 0; c < Cc; ++c) {
        const float* xc = xb + c * HW;
        const float* wk = wc + c * 9;
#pragma unroll
        for (int i = 0; i < 3; ++i) {
            int y = h - 1 + i;
            bool yok = (y >= 0) && (y < Hh);
#pragma unroll
            for (int j = 0; j < 3; ++j) {
                int xx = w - 1 + j;
                bool ok = yok && (xx >= 0) && (xx < Ww);
                float xv = ok ? xc[y * Ww + xx] : 0.0f;
                acc = fmaf(xv, wk[i * 3 + j], acc);
            }
        }
    }
    offs[idx] = acc;
}

// ---------------------------------------------------------------------------
// Kernel 2: bilinear gather into LDS, then fp32 WMMA GEMM
//   out[oc, pixel] += W[oc, c*9+k] * Vals[c*9+k, pixel]
// 128 threads = 4 wave32; wave i owns oc-tile [16i, 16i+16), 16 pixels wide.
// ---------------------------------------------------------------------------
__global__ __launch_bounds__(128) void deform_wmma_kernel(
    const float* __restrict__ x, const float* __restrict__ offs,
    const float* __restrict__ w_def, float* __restrict__ out)
{
    __shared__ float Vals[KC * TP];   // 36 KB: row = c*9+k, col = local pixel

    int tid = threadIdx.x;

    // ---------------- Phase A: bilinear im2col into LDS ----------------
    {
        int p  = tid >> 3;           // local pixel 0..15
        int cg = tid & 7;            // channel group (c = cg + 8*i)
        int g  = blockIdx.x * TP + p;
        int b  = g >> 14;            // / HW
        int hw = g & (HW - 1);
        int h = hw >> 7, w = hw & 127;

        const float* xb = x + (size_t)b * Cc * HW;
        const float* ob = offs + (size_t)b * NOFF * HW + hw;

        for (int k = 0; k < KKk; ++k) {
            float oy = ob[(2 * k) * HW];
            float ox = ob[(2 * k + 1) * HW];
            float py = oy + (float)(h - 1 + k / 3);
            float px = ox + (float)(w - 1 + k % 3);

            float fy = floorf(py), fx = floorf(px);
            int   y0 = (int)fy,    x0 = (int)fx;
            float wy1 = py - fy,   wx1 = px - fx;
            float wy0 = 1.0f - wy1, wx0 = 1.0f - wx1;
            int y1 = y0 + 1, x1 = x0 + 1;

            bool vy0 = (y0 >= 0) && (y0 < Hh), vy1 = (y1 >= 0) && (y1 < Hh);
            bool vx0 = (x0 >= 0) && (x0 < Ww), vx1 = (x1 >= 0) && (x1 < Ww);
            float w00 = (vy0 && vx0) ? wy0 * wx0 : 0.0f;
            float w01 = (vy0 && vx1) ? wy0 * wx1 : 0.0f;
            float w10 = (vy1 && vx0) ? wy1 * wx0 : 0.0f;
            float w11 = (vy1 && vx1) ? wy1 * wx1 : 0.0f;

            int y0c = min(max(y0, 0), Hh - 1), y1c = min(max(y1, 0), Hh - 1);
            int x0c = min(max(x0, 0), Ww - 1), x1c = min(max(x1, 0), Ww - 1);
            int i00 = y0c * Ww + x0c, i01 = y0c * Ww + x1c;
            int i10 = y1c * Ww + x0c, i11 = y1c * Ww + x1c;

#pragma unroll
            for (int ci = 0; ci < 8; ++ci) {
                int c = cg + ci * 8;
                const float* xc = xb + c * HW;
                float v = w00 * xc[i00] + w01 * xc[i01]
                        + w10 * xc[i10] + w11 * xc[i11];
                Vals[(c * KKk + k) * TP + p] = v;
            }
        }
    }
    __syncthreads();

    // ---------------- Phase B: fp32 WMMA GEMM ----------------
    int wave = tid >> 5;
    int lane = tid & 31;
    int row  = lane & 15;            // A: M row / B: N col
    int hi   = lane >> 4;            // 0: K 0,1   1: K 2,3
    int m_base = wave * 16;
    int oc = m_base + row;

    const float* arow = w_def + oc * KC;   // contiguous: row index == c*9+k

    v8f acc = {};
    for (int kb = 0; kb < KC; kb += 4) {
        int k0 = kb + hi * 2;
        // A: 16x4 tile of weights (8-byte aligned contiguous pair)
        v2f a = *reinterpret_cast<const v2f*>(arow + k0);
        // B: 4x16 tile of gathered values from LDS
        v2f bb;
        bb.x = Vals[k0 * TP + row];
        bb.y = Vals[(k0 + 1) * TP + row];
        acc = __builtin_amdgcn_wmma_f32_16x16x4_f32(
            /*neg_a=*/false, a, /*neg_b=*/false, bb,
            /*c_mod=*/(short)0, acc, /*reuse_a=*/false, /*reuse_b=*/false);
    }

    // ---------------- Store: C layout VGPR r -> M = r (lo half) / r+8 (hi) --
    int g  = blockIdx.x * TP + row;
    int b  = g >> 14;
    int hw = g & (HW - 1);
#pragma unroll
    for (int r = 0; r < 8; ++r) {
        int occ = m_base + r + hi * 8;
        out[((size_t)b * Cc + occ) * HW + hw] = acc[r];
    }
}

// ---------------------------------------------------------------------------
extern "C" void kernel_launch(void* const* d_in, const int* in_sizes, int n_in,
                              void* d_out, int out_size, void* d_ws, size_t ws_size,
                              hipStream_t stream)
{
    const float* x     = (const float*)d_in[0];
    const float* w_off = (const float*)d_in[1];
    const float* b_off = (const float*)d_in[2];
    const float* w_def = (const float*)d_in[3];
    float* out  = (float*)d_out;
    float* offs = (float*)d_ws;          // B*18*H*W floats = 9.4 MB scratch

    int n_off = Bn * NOFF * HW;          // 2,359,296 threads, 256/block
    offset_conv_kernel<<<n_off / 256, 256, 0, stream>>>(x, w_off, b_off, offs);

    int n_wg = (Bn * HW) / TP;           // 8192 workgroups
    deform_wmma_kernel<<<n_wg, 128, 0, stream>>>(x, offs, w_def, out);
}